// OpeningLoss2D_20916490731956
// MI455X (gfx1250) — compile-verified
//
#include <hip/hip_runtime.h>
#include <stdint.h>

#define H 512
#define W 512
#define ROWS 16                     // output rows per block
#define TROWS (ROWS + 2)            // tile rows incl. top/bottom halo
#define BLK 256                     // 8 waves (wave32)
#define RB_PER_CH (H / ROWS)        // 32 row-blocks per channel
#define NCHANNELS (16 * 8)          // 128
#define NBLOCKS (NCHANNELS * RB_PER_CH)  // 4096
#define NELEM (16.0f * 8.0f * 512.0f * 512.0f)

typedef __attribute__((ext_vector_type(2))) float v2f;
typedef __attribute__((ext_vector_type(8))) float v8f;

__global__ __launch_bounds__(BLK) void opening_mse_partial(
    const float* __restrict__ labels, float* __restrict__ partial) {
  __shared__ float tile[TROWS * W];
  __shared__ float swave[BLK / 32];

  const int tid = threadIdx.x;
  const int bid = blockIdx.x;
  const int ch  = bid >> 5;                 // /RB_PER_CH
  const int rb  = bid & (RB_PER_CH - 1);
  const int r0  = rb * ROWS;
  const float* __restrict__ src = labels + (size_t)ch * (H * W);

  // ---- Stage 18 rows x 512 cols into LDS via CDNA5 async global->LDS ----
  // 18 rows * 128 float4 = 2304 vector copies, 9 per thread, 16B aligned.
  #pragma unroll
  for (int it = 0; it < (TROWS * (W / 4)) / BLK; ++it) {
    int idx  = tid + it * BLK;
    int lrow = idx >> 7;                    // /128 float4 per row
    int c4   = idx & 127;
    int grow = r0 - 1 + lrow;               // global row (edge-replicate clamp)
    grow = grow < 0 ? 0 : (grow > H - 1 ? H - 1 : grow);
    const float* gp = src + grow * W + c4 * 4;
    unsigned lds_off = (unsigned)(uintptr_t)&tile[lrow * W + c4 * 4];
    asm volatile("global_load_async_to_lds_b128 %0, %1, off"
                 :: "v"(lds_off), "v"((unsigned long long)(uintptr_t)gp)
                 : "memory");
  }
  asm volatile("s_wait_asynccnt 0" ::: "memory");
  __syncthreads();

  // ---- 2x2 grey opening + squared-diff accumulation ----
  // er[i][j]   = min(x[i-1..i][j-1..j])  (clamped at input, top/left pad)
  // op[i][j]   = max(er[i..i+1][j..j+1]) (clamped at er, bottom/right pad)
  float acc = 0.0f;
  #pragma unroll 4
  for (int it = 0; it < (ROWS * W) / BLK; ++it) {
    int idx  = tid + it * BLK;              // 0..8191
    int orow = idx >> 9;                    // 0..15
    int j    = idx & (W - 1);
    int gi   = r0 + orow;                   // global output row
    int jm   = j ? j - 1 : 0;
    const float* row0 = &tile[orow * W];          // row gi-1 (clamped)
    const float* row1 = &tile[(orow + 1) * W];    // row gi
    const float* row2 = &tile[(orow + 2) * W];    // row gi+1 (clamped)

    float x00 = row0[jm], x01 = row0[j];
    float x10 = row1[jm], x11 = row1[j];
    float x20 = row2[jm], x21 = row2[j];
    float cT0 = fminf(x00, x10), cT1 = fminf(x01, x11);
    float cB0 = fminf(x10, x20), cB1 = fminf(x11, x21);
    float e00 = fminf(cT0, cT1);            // er[gi][j]
    float e10 = fminf(cB0, cB1);            // er[gi+1][j]

    int jp = (j < W - 1) ? j + 1 : j;
    float x02 = row0[jp], x12 = row1[jp], x22 = row2[jp];
    float cT2 = fminf(x02, x12), cB2 = fminf(x12, x22);
    float e01 = (j < W - 1) ? fminf(cT1, cT2) : e00;  // er[gi][j+1] clamped
    float e11 = (j < W - 1) ? fminf(cB1, cB2) : e10;  // er[gi+1][j+1] clamped
    if (gi == H - 1) { e10 = e00; e11 = e01; }        // er row clamp at bottom

    float op = fmaxf(fmaxf(e00, e01), fmaxf(e10, e11));
    float d  = x11 - op;
    acc = fmaf(d, d, acc);
  }

  // ---- Per-wave reduction via V_WMMA_F32_16X16X4_F32 ----
  // A = 16x4 matrix holding the wave's 32 lane sums (other K slots zero),
  // B = ones(4x16)  =>  D[m][n] = sum_k A[m][k].  Summing D's 8 VGPRs per
  // lane then adding lane0+lane16 yields the full 32-lane sum.
  v2f A;  A[0] = acc;  A[1] = 0.0f;
  v2f Bv; Bv[0] = 1.0f; Bv[1] = 1.0f;
  v8f Cz = {};
  v8f D = __builtin_amdgcn_wmma_f32_16x16x4_f32(false, A, false, Bv,
                                                (short)0, Cz, false, false);
  float t = ((D[0] + D[1]) + (D[2] + D[3])) + ((D[4] + D[5]) + (D[6] + D[7]));
  float wsum = __int_as_float(__builtin_amdgcn_readlane(__float_as_int(t), 0)) +
               __int_as_float(__builtin_amdgcn_readlane(__float_as_int(t), 16));
  if ((tid & 31) == 0) swave[tid >> 5] = wsum;
  __syncthreads();
  if (tid == 0) {
    float s = 0.0f;
    #pragma unroll
    for (int w = 0; w < BLK / 32; ++w) s += swave[w];
    partial[bid] = s;
  }
}

__global__ __launch_bounds__(256) void opening_mse_final(
    const float* __restrict__ partial, float* __restrict__ out) {
  __shared__ float s[256];
  float a = 0.0f;
  for (int i = threadIdx.x; i < NBLOCKS; i += 256) a += partial[i];
  s[threadIdx.x] = a;
  __syncthreads();
  #pragma unroll
  for (int w = 128; w > 0; w >>= 1) {
    if (threadIdx.x < w) s[threadIdx.x] += s[threadIdx.x + w];
    __syncthreads();
  }
  if (threadIdx.x == 0) out[0] = s[0] * (1.0f / NELEM);  // * 2^-25, exact
}

extern "C" void kernel_launch(void* const* d_in, const int* in_sizes, int n_in,
                              void* d_out, int out_size, void* d_ws, size_t ws_size,
                              hipStream_t stream) {
  (void)in_sizes; (void)n_in; (void)out_size; (void)ws_size;
  const float* labels = (const float*)d_in[0];
  float* partial = (float*)d_ws;            // NBLOCKS floats (16 KB)
  float* out = (float*)d_out;               // single f32 scalar
  opening_mse_partial<<<NBLOCKS, BLK, 0, stream>>>(labels, partial);
  opening_mse_final<<<1, 256, 0, stream>>>(partial, out);
}